// Local_Ranking_Loss_47313359733321
// MI455X (gfx1250) — compile-verified
//
#include <hip/hip_runtime.h>
#include <stdint.h>

// ---------------- problem constants (fixed by the reference) ----------------
#define RADK  5
#define DILK  6
#define TOPK  8
#define KSZ   11            // kernel size
#define NTAP  121           // KSZ*KSZ taps
#define PADR  30            // RADK*DILK

#define BATCH 2
#define HH    256
#define WW    512
#define TOTAL_TERMS 2097152 // BATCH*HH*WW*TOPK

// ---------------- tiling ----------------
#define TW 16
#define THT 16
#define HALO_W (TW  + 2*PADR)   // 76
#define HALO_H (THT + 2*PADR)   // 76
#define LSTR   80               // padded LDS row stride (floats): rows 80 apart -> +16 banks, conflict-free
#define CELLS  (HALO_H * HALO_W)  // 5776 cells actually used per array
#define ARRF   (HALO_H * LSTR)    // 6080 floats reserved per array
#define NBLK   (BATCH * (HH/THT) * (WW/TW))   // 1024 workgroups

#define LN2F 0.69314718055994530942f

// ---------------- CDNA5 async global->LDS staging ----------------
// Per-lane async copy of one float from the global address gsrc to the LDS address ldst, tracked on ASYNCcnt (gfx1250 path).
// Builtin signature (from hipcc diagnostic): (AS1 int*, AS3 int*, imm offset, imm cpol).
__device__ __forceinline__ void async_copy_b32(const float* gsrc, float* ldst) {
#if __has_builtin(__builtin_amdgcn_global_load_async_to_lds_b32)
  __builtin_amdgcn_global_load_async_to_lds_b32(
      (__attribute__((address_space(1))) int*)(void*)(gsrc),
      (__attribute__((address_space(3))) int*)(void*)(ldst),
      /*offset=*/0, /*cpol=*/0);
#else
  *ldst = *gsrc;
#endif
}

__device__ __forceinline__ void wait_async_done() {
#if __has_builtin(__builtin_amdgcn_s_wait_asynccnt)
  __builtin_amdgcn_s_wait_asynccnt(0);
#else
  asm volatile("s_wait_asynccnt 0" ::: "memory");
#endif
}

// log1p(x) for x >= 0 without libm: ln2 * v_log_f32(1+x)
__device__ __forceinline__ float log1p_hw(float x) {
#if __has_builtin(__builtin_amdgcn_logf)
  return LN2F * __builtin_amdgcn_logf(1.0f + x);
#else
  return LN2F * __log2f(1.0f + x);
#endif
}

// ---------------- main kernel: one pixel per thread, 16x16 tile per block ----------------
extern "C" __global__ __launch_bounds__(256)
void lrl_main(const float* __restrict__ disp,
              const float* __restrict__ depth,
              const float* __restrict__ vote,
              float* __restrict__ partial)
{
  extern __shared__ float smem[];
  float* sD  = smem;               // depth halo  (edge-replicated via clamped addresses)
  float* sS  = smem + ARRF;        // disp  halo  (edge-replicated)
  float* sV  = smem + 2 * ARRF;    // vote  halo  (constant-1.0 padded, fixed up below)
  float* red = smem + 3 * ARRF;    // 256 reduction slots

  const int tx  = threadIdx.x, ty = threadIdx.y;
  const int tid = ty * TW + tx;
  const int gx0 = blockIdx.x * TW;
  const int gy0 = blockIdx.y * THT;
  const size_t ibase = (size_t)blockIdx.z * (HH * WW);

  // ---- stage halos with async-to-LDS copies; clamp = edge replicate ----
  for (int i = tid; i < CELLS; i += 256) {
    int r  = i / HALO_W;
    int c  = i - r * HALO_W;
    int iy = gy0 - PADR + r;
    int ix = gx0 - PADR + c;
    int cy = min(max(iy, 0), HH - 1);
    int cx = min(max(ix, 0), WW - 1);
    size_t g = ibase + (size_t)cy * WW + cx;
    int    l = r * LSTR + c;
    async_copy_b32(depth + g, sD + l);
    async_copy_b32(disp  + g, sS + l);
    async_copy_b32(vote  + g, sV + l);
  }
  wait_async_done();   // this wave's async writes have landed in LDS

  // ---- vote uses constant padding = 1.0 (>= any real vote in [0,1)) ----
  for (int i = tid; i < CELLS; i += 256) {
    int r  = i / HALO_W;
    int c  = i - r * HALO_W;
    int iy = gy0 - PADR + r;
    int ix = gx0 - PADR + c;
    if (iy < 0 || iy >= HH || ix < 0 || ix >= WW)
      sV[r * LSTR + c] = 1.0f;
  }
  __syncthreads();

  // ---- top-8 smallest votes of the 121 dilated taps ----
  // votes are non-negative floats -> uint ordering == float ordering.
  // pack the 7-bit tap id into the low mantissa bits => single-array min/max bubble.
  unsigned best[TOPK];
#pragma unroll
  for (int j = 0; j < TOPK; ++j) best[j] = 0xFFFFFFFFu;

  const int lbase = ty * LSTR + tx;   // tap (ky,kx) lives at lbase + ky*6*LSTR + kx*6
#pragma unroll
  for (int ky = 0; ky < KSZ; ++ky) {
#pragma unroll
    for (int kx = 0; kx < KSZ; ++kx) {
      float v = sV[lbase + ky * (DILK * LSTR) + kx * DILK];
      unsigned key = (__float_as_uint(v) & 0xFFFFFF80u) | (unsigned)(ky * KSZ + kx);
#pragma unroll
      for (int j = 0; j < TOPK; ++j) {          // branchless sorted insert
        unsigned lo = key < best[j] ? key : best[j];
        unsigned hi = key < best[j] ? best[j] : key;
        best[j] = lo;
        key     = hi;
      }
    }
  }

  // ---- evaluate the 8 selected ranking terms ----
  const int  cofs = lbase + PADR * LSTR + PADR;   // center cell in halo
  const float dc = sD[cofs];
  const float sc = sS[cofs];

  float acc = 0.0f;
#pragma unroll
  for (int j = 0; j < TOPK; ++j) {
    int idx = (int)(best[j] & 127u);
    int ky  = idx / KSZ;
    int kx  = idx - ky * KSZ;
    int off = lbase + ky * (DILK * LSTR) + kx * DILK;
    float dg = dc - sD[off];
    float sg = sc - sS[off];
    float dd = sg / (fabsf(sg) + 1e-8f);                // softsign-ish direction
    float t  = fmaxf(fabsf(dg) - 1.0f, 0.0f);           // softshrink(|dg|,1)
    float dw = __builtin_copysignf(t / (1.0f + t), dg); // softsign(softshrink)
    float dv = log1p_hw(sg * sg);                       // log1p(sg^2) via v_log_f32
    acc += fmaxf(-(dw * dd * dv), 0.0f);
  }

  // ---- deterministic in-block tree reduction ----
  red[tid] = acc;
  __syncthreads();
#pragma unroll
  for (int s = 128; s > 0; s >>= 1) {
    if (tid < s) red[tid] += red[tid + s];
    __syncthreads();
  }
  if (tid == 0) {
    int pidx = (blockIdx.z * (HH / THT) + blockIdx.y) * (WW / TW) + blockIdx.x;
    partial[pidx] = red[0];
  }
}

// ---------------- final reduce: 1024 partials -> mean ----------------
extern "C" __global__ __launch_bounds__(256)
void lrl_reduce(const float* __restrict__ partial, float* __restrict__ out)
{
  __shared__ float red[256];
  int tid = threadIdx.x;
  float s = 0.0f;
  for (int i = tid; i < NBLK; i += 256) s += partial[i];
  red[tid] = s;
  __syncthreads();
#pragma unroll
  for (int k = 128; k > 0; k >>= 1) {
    if (tid < k) red[tid] += red[tid + k];
    __syncthreads();
  }
  if (tid == 0) out[0] = red[0] * (1.0f / (float)TOTAL_TERMS);
}

// ---------------- host entry ----------------
extern "C" void kernel_launch(void* const* d_in, const int* in_sizes, int n_in,
                              void* d_out, int out_size, void* d_ws, size_t ws_size,
                              hipStream_t stream) {
  (void)in_sizes; (void)n_in; (void)out_size; (void)ws_size;
  const float* disp  = (const float*)d_in[0];
  const float* depth = (const float*)d_in[1];
  const float* vote  = (const float*)d_in[2];
  float* partial = (float*)d_ws;                 // 1024 floats of scratch

  dim3 grid(WW / TW, HH / THT, BATCH);           // 32 x 16 x 2 = 1024 blocks
  dim3 block(TW, THT, 1);                        // 256 threads = 8 wave32 waves
  size_t shmem = (size_t)(3 * ARRF + 256) * sizeof(float);  // ~74 KB (<< 320 KB/WGP)

  hipLaunchKernelGGL(lrl_main, grid, block, shmem, stream, disp, depth, vote, partial);
  hipLaunchKernelGGL(lrl_reduce, dim3(1), dim3(256), 0, stream, partial, (float*)d_out);
}